// TopKAbsolutes2D_27504970563634
// MI455X (gfx1250) — compile-verified
//
#include <hip/hip_runtime.h>
#include <stdint.h>

// Problem constants (from reference): B=64, C=64, H=W=128 -> N = 2^20 per row.
#define ROWS 64
#define ROW_SHIFT 20
#define N_PER_ROW (1 << ROW_SHIFT)

#define HBITS 14
#define HBINS (1 << HBITS)
#define TOPSHIFT (31 - HBITS)

#define BLOCKS_PER_ROW 32
#define K1_THREADS 256
#define CAP 4096

// Native clang vector type: required for __builtin_nontemporal_* and lowers to b128 ops.
typedef float vfloat4 __attribute__((ext_vector_type(4)));

// ---------------------------------------------------------------------------
// gfx1250 async global->LDS copy (ISA §15.18.3 opcode 98) + split wait counters
// ---------------------------------------------------------------------------
__device__ __forceinline__ void async_load_b128(uint32_t lds_off, const void* gaddr) {
    asm volatile("global_load_async_to_lds_b128 %0, %1, off"
                 :
                 : "v"(lds_off), "v"((uint64_t)(uintptr_t)gaddr)
                 : "memory");
}
__device__ __forceinline__ void wait_asynccnt0() {
    asm volatile("s_wait_asynccnt 0x0" ::: "memory");
}
__device__ __forceinline__ void wait_asynccnt1() {
    asm volatile("s_wait_asynccnt 0x1" ::: "memory");
}
__device__ __forceinline__ void wait_dscnt0() {
    asm volatile("s_wait_dscnt 0x0" ::: "memory");
}

__device__ __forceinline__ uint32_t lds_offset_of(const void* p) {
    // Generic LDS addresses carry the LDS byte offset in addr[31:0] (ISA §10.2).
    return (uint32_t)(uintptr_t)p;
}

// ---------------------------------------------------------------------------
// Kernel 1: per-row histogram of top-14 bits of |x| bit pattern.
// Streams data through LDS with double-buffered async loads (ASYNCcnt).
// Regular-temporal loads: x should stay resident in the 192 MB L2 for pass 2.
// ---------------------------------------------------------------------------
__global__ void hist_kernel(const float* __restrict__ x, int* __restrict__ ghist) {
    __shared__ int lh[HBINS];                  // 64 KB private histogram (320 KB/WGP budget)
    __shared__ float stage[2][K1_THREADS * 4]; // 8 KB async staging

    const int tid = threadIdx.x;
    for (int i = tid; i < HBINS; i += K1_THREADS) lh[i] = 0;
    __syncthreads();

    const int row = blockIdx.x / BLOCKS_PER_ROW;
    const int blk = blockIdx.x % BLOCKS_PER_ROW;
    const int seg_elems = N_PER_ROW / BLOCKS_PER_ROW;                 // 32768
    const float* src = x + ((long long)row << ROW_SHIFT) + (long long)blk * seg_elems;
    const int iters = seg_elems / (K1_THREADS * 4);                   // 32

    const uint32_t lds0 = lds_offset_of(&stage[0][tid * 4]);
    const uint32_t lds1 = lds_offset_of(&stage[1][tid * 4]);

    // Prime pipeline: tile 0 -> buffer 0
    async_load_b128(lds0, src + tid * 4);

    for (int it = 0; it < iters; ++it) {
        const int buf = it & 1;
        if (it + 1 < iters) {
            // buffer being overwritten was fully consumed last iteration;
            // make sure those DS reads (and histogram atomics) retired.
            wait_dscnt0();
            async_load_b128(buf ? lds0 : lds1,
                            src + (long long)(it + 1) * (K1_THREADS * 4) + tid * 4);
            wait_asynccnt1(); // tile `it` has landed in LDS
        } else {
            wait_asynccnt0();
        }
        const float* sp = &stage[buf][tid * 4];
#pragma unroll
        for (int j = 0; j < 4; ++j) {
            uint32_t ab = __float_as_uint(sp[j]) & 0x7fffffffu;
            atomicAdd(&lh[ab >> TOPSHIFT], 1);
        }
    }
    __syncthreads();

    int* gh = ghist + (long long)row * HBINS;
    for (int i = tid; i < HBINS; i += K1_THREADS) {
        int v = lh[i];
        if (v) atomicAdd(&gh[i], v);
    }
}

// ---------------------------------------------------------------------------
// Kernel 2: one block per row; scan histogram from the top to find the bin
// containing the k-th largest |x| and the count of strictly greater elements.
// ---------------------------------------------------------------------------
__global__ void bucket_kernel(const int* __restrict__ ghist, const int* __restrict__ topk,
                              int* __restrict__ bucket, int* __restrict__ cntG) {
    __shared__ int s_part[256];
    __shared__ int s_before[256];
    const int row = blockIdx.x;
    const int t = threadIdx.x;
    const int k = *topk;
    const int* h = ghist + (long long)row * HBINS;
    const int chunk = HBINS / 256; // 64 bins, descending chunks
    const int lo = HBINS - (t + 1) * chunk;

    int s = 0;
    for (int b = 0; b < chunk; ++b) s += h[lo + b];
    s_part[t] = s;
    __syncthreads();

    if (t == 0) {
        int acc = 0;
        for (int i = 0; i < 256; ++i) { s_before[i] = acc; acc += s_part[i]; }
    }
    __syncthreads();

    const int before = s_before[t];
    if (before < k && before + s_part[t] >= k) {
        int c = before;
        for (int b = HBINS - t * chunk - 1; b >= lo; --b) {
            const int hv = h[b];
            if (c + hv >= k) { bucket[row] = b; cntG[row] = c; break; }
            c += hv;
        }
    }
}

// ---------------------------------------------------------------------------
// Kernel 3: streaming scatter. >bucket -> keep, <bucket -> 0,
// ==bucket -> 0 + push candidate (resolved by kernel 4).
// Last read of x -> non-temporal load; pure output stream -> non-temporal
// store so 256 MB of writes don't evict x/metadata from the 192 MB L2.
// ---------------------------------------------------------------------------
__global__ void scatter_kernel(const float* __restrict__ x, float* __restrict__ out,
                               const int* __restrict__ bucket,
                               int* __restrict__ candCnt, float* __restrict__ candVal,
                               int* __restrict__ candIdx) {
    const long long g4 = (long long)blockIdx.x * blockDim.x + threadIdx.x; // float4 id
    const long long e0 = g4 * 4;
    const int row = (int)(e0 >> ROW_SHIFT); // a 4-vector never crosses a row (N % 4 == 0)
    const int t = bucket[row];

    const vfloat4 v = __builtin_nontemporal_load(&((const vfloat4*)x)[g4]);
    vfloat4 o;
#pragma unroll
    for (int j = 0; j < 4; ++j) {
        const uint32_t ab = __float_as_uint(v[j]) & 0x7fffffffu;
        const int top = (int)(ab >> TOPSHIFT);
        if (top > t) {
            o[j] = v[j];
        } else {
            o[j] = 0.0f;
            if (top == t) {
                const int pos = atomicAdd(&candCnt[row], 1);
                if (pos < CAP) {
                    candVal[row * CAP + pos] = v[j];
                    candIdx[row * CAP + pos] = (int)(e0 + j - ((long long)row << ROW_SHIFT));
                }
            }
        }
    }
    __builtin_nontemporal_store(o, &((vfloat4*)out)[g4]);
}

// ---------------------------------------------------------------------------
// Kernel 4: per-row, rank candidates in LDS (ties -> lowest index, matching
// jax.lax.top_k) and scatter the remaining m = k - cntGreater winners.
// ---------------------------------------------------------------------------
__global__ void rank_kernel(float* __restrict__ out, const int* __restrict__ topk,
                            const int* __restrict__ cntG, const int* __restrict__ candCnt,
                            const float* __restrict__ candVal, const int* __restrict__ candIdx) {
    __shared__ uint32_t sAb[CAP];
    __shared__ int sIdx[CAP];
    const int row = blockIdx.x;
    const int tid = threadIdx.x;

    int c = candCnt[row];
    if (c > CAP) c = CAP;
    int m = *topk - cntG[row];
    if (m > c) m = c;
    if (m <= 0) return; // uniform across block

    for (int i = tid; i < c; i += blockDim.x) {
        sAb[i] = __float_as_uint(candVal[row * CAP + i]) & 0x7fffffffu;
        sIdx[i] = candIdx[row * CAP + i];
    }
    __syncthreads();

    for (int i = tid; i < c; i += blockDim.x) {
        const uint32_t ai = sAb[i];
        const int ii = sIdx[i];
        int rank = 0;
        for (int j = 0; j < c; ++j) {
            const uint32_t aj = sAb[j];
            rank += (aj > ai) || (aj == ai && sIdx[j] < ii);
        }
        if (rank < m) out[((long long)row << ROW_SHIFT) + ii] = candVal[row * CAP + i];
    }
}

// ---------------------------------------------------------------------------
extern "C" void kernel_launch(void* const* d_in, const int* in_sizes, int n_in,
                              void* d_out, int out_size, void* d_ws, size_t ws_size,
                              hipStream_t stream) {
    const float* x = (const float*)d_in[0];
    const int* topk = (const int*)d_in[1];
    float* out = (float*)d_out;

    // Workspace layout (~6.2 MB total)
    int* hist = (int*)d_ws;                       // 64 * 16384 ints   = 4 MB
    int* bucket = hist + (long long)ROWS * HBINS; // 64 ints
    int* cntG = bucket + ROWS;                    // 64 ints
    int* candCnt = cntG + ROWS;                   // 64 ints
    float* candVal = (float*)(candCnt + ROWS);    // 64 * 4096 floats  = 1 MB
    int* candIdx = (int*)(candVal + (long long)ROWS * CAP); // 1 MB

    // Zero histogram + metadata every call (graph-capturable stream memset)
    (void)hipMemsetAsync(d_ws, 0, ((size_t)ROWS * HBINS + 3 * ROWS) * sizeof(int), stream);

    hist_kernel<<<ROWS * BLOCKS_PER_ROW, K1_THREADS, 0, stream>>>(x, hist);
    bucket_kernel<<<ROWS, 256, 0, stream>>>(hist, topk, bucket, cntG);

    const long long nf4 = ((long long)ROWS << ROW_SHIFT) / 4; // 16,777,216 vfloat4s
    scatter_kernel<<<(int)(nf4 / 256), 256, 0, stream>>>(x, out, bucket, candCnt, candVal, candIdx);
    rank_kernel<<<ROWS, 256, 0, stream>>>(out, topk, cntG, candCnt, candVal, candIdx);
}